// LocalClusterReshapeFromNeighbours_38285338476783
// MI455X (gfx1250) — compile-verified
//
#include <hip/hip_runtime.h>
#include <hip/hip_bf16.h>
#include <stdint.h>

// Problem constants (from the reference file).
#define V_TOT   200000
#define F_DIM   128
#define K_IN    64
#define K_SEL   16
#define V_SEL   50000

#define WAVES_PER_BLOCK 8
#define BLOCK_THREADS   (WAVES_PER_BLOCK * 32)

typedef float v4f __attribute__((ext_vector_type(4)));
typedef int   v4i __attribute__((ext_vector_type(4)));

// ---- CDNA5 async global<->LDS path (gfx1250), with safe fallback ---------
#if defined(__gfx1250__) && \
    __has_builtin(__builtin_amdgcn_global_load_async_to_lds_b128) && \
    __has_builtin(__builtin_amdgcn_global_store_async_from_lds_b128)
#define USE_ASYNC_LDS 1
#else
#define USE_ASYNC_LDS 0
#endif

// gfx12-family CPol: TH in bits [2:0]; TH_STORE_NT == 1. Non-temporal
// stores keep the 410 MB output stream from evicting the L2-resident
// 102 MB feature table.
#define CPOL_RT       0
#define CPOL_STORE_NT 1

#if USE_ASYNC_LDS
// Builtin signature (from clang diagnostic): first param is int4* in AS1
// (printed as "int4 __device__ *"), second is the LDS-side pointer (AS3).
typedef __attribute__((address_space(1))) v4i as1_v4i;
typedef __attribute__((address_space(3))) v4i as3_v4i;

__device__ __forceinline__ void wait_async0() {
#if __has_builtin(__builtin_amdgcn_s_wait_asynccnt)
    __builtin_amdgcn_s_wait_asynccnt(0);
#else
    asm volatile("s_wait_asynccnt 0" ::: "memory");
#endif
}
#endif

__device__ __forceinline__ unsigned long long umin64(unsigned long long a,
                                                     unsigned long long b) {
    return a < b ? a : b;
}

// One wave32 per selected row:
//   1) stable top-16-by-distance over 64 (distance, nidx) candidates
//   2) gather 16 feature rows (128 f32 each) into out[s, k*128 .. ]
__global__ __launch_bounds__(BLOCK_THREADS)
void knn_topk_gather_kernel(const float* __restrict__ feat,      // [V, F]
                            const float* __restrict__ dist,      // [V, K_IN]
                            const int*   __restrict__ nidx,      // [V, K_IN]
                            const int*   __restrict__ sel,       // [V_SEL]
                            float*       __restrict__ out)       // [V_SEL, K_SEL*F]
{
#if USE_ASYNC_LDS
    __shared__ float stage[WAVES_PER_BLOCK][K_SEL * F_DIM];      // 8 KB per wave
#endif
    const int lane = threadIdx.x & 31;
    const int wave = threadIdx.x >> 5;
    const int s    = blockIdx.x * WAVES_PER_BLOCK + wave;
    if (s >= V_SEL) return;

    const int v = sel[s];

    // ---- load 64 candidates, 2 per lane --------------------------------
    const int c0 = lane;
    const int c1 = lane + 32;
    const size_t rb = (size_t)v * K_IN;
    int   i0 = nidx[rb + c0];
    int   i1 = nidx[rb + c1];
    float d0 = dist[rb + c0];
    float d1 = dist[rb + c1];
    if (i0 < 0) d0 = 1.0e9f;   // BIG sentinel, same as reference
    if (i1 < 0) d1 = 1.0e9f;

    // Non-negative floats: IEEE bit pattern is order-preserving.
    // Embed the column for argsort-stable tie-breaking.
    unsigned long long k0 =
        ((unsigned long long)__float_as_uint(d0) << 8) | (unsigned)c0;
    unsigned long long k1 =
        ((unsigned long long)__float_as_uint(d1) << 8) | (unsigned)c1;

    // ---- 16 rounds of wave-wide min selection --------------------------
    int knn[K_SEL];
#pragma unroll
    for (int k = 0; k < K_SEL; ++k) {
        unsigned long long m = umin64(k0, k1);
#pragma unroll
        for (int off = 16; off > 0; off >>= 1)
            m = umin64(m, __shfl_xor(m, off, 32));

        const int col = (int)(m & 0xFF);        // winning column (uniform)
        const int src = col & 31;               // lane owning it
        knn[k] = __shfl(col < 32 ? i0 : i1, src, 32);
        if (lane == src) {                      // retire the winner
            if (col < 32) k0 = ~0ull; else k1 = ~0ull;
        }
    }

    // ---- gather: 16 rows x 512 B each ----------------------------------
    const size_t obase = (size_t)s * (K_SEL * F_DIM);

#if USE_ASYNC_LDS
    float* stg = &stage[wave][0];
    // Batch 1: fire all async loads (mem -> LDS), 16 B per lane each.
    // cpol=RT so gathered feature lines stay resident in the 192 MB L2.
#pragma unroll
    for (int k = 0; k < K_SEL; ++k) {
        const int nid = knn[k];                 // wave-uniform
        if (nid >= 0) {
            const float* gsrc = feat + (size_t)nid * F_DIM + lane * 4;
            float*       ldst = stg + k * F_DIM + lane * 4;
            __builtin_amdgcn_global_load_async_to_lds_b128(
                (as1_v4i*)gsrc, (as3_v4i*)ldst, 0, CPOL_RT);
        }
    }
    wait_async0();
    // Batch 2: fire all async stores (LDS -> mem) with TH_NT (streaming
    // write-once output); zeros for invalid rows.
#pragma unroll
    for (int k = 0; k < K_SEL; ++k) {
        const int nid = knn[k];
        float* gdst = out + obase + k * F_DIM + lane * 4;
        if (nid >= 0) {
            float* lsrc = stg + k * F_DIM + lane * 4;
            __builtin_amdgcn_global_store_async_from_lds_b128(
                (as1_v4i*)gdst, (as3_v4i*)lsrc, 0, CPOL_STORE_NT);
        } else {
            v4f z = (v4f)0.0f;
            __builtin_nontemporal_store(z, (v4f*)gdst);
        }
    }
    wait_async0();
#else
    // Fallback: b128 load -> non-temporal b128 store (don't evict the
    // L2-resident 102 MB feature table with the 410 MB streaming output).
#pragma unroll
    for (int k = 0; k < K_SEL; ++k) {
        const int nid = knn[k];
        v4f val = (v4f)0.0f;
        if (nid >= 0)
            val = ((const v4f*)(feat + (size_t)nid * F_DIM))[lane];
        __builtin_nontemporal_store(val, (v4f*)(out + obase + k * F_DIM) + lane);
    }
#endif
}

extern "C" void kernel_launch(void* const* d_in, const int* in_sizes, int n_in,
                              void* d_out, int out_size, void* d_ws, size_t ws_size,
                              hipStream_t stream) {
    const float* feat = (const float*)d_in[0];   // features [V, F]
    const float* dist = (const float*)d_in[1];   // distances [V, K_IN]
    const int*   nid  = (const int*)  d_in[2];   // nidx [V, K_IN]
    const int*   sel  = (const int*)  d_in[3];   // sel_idx [V_SEL, 1]
    // d_in[4] is the python scalar K (==16), baked in as K_SEL.
    float* out = (float*)d_out;                  // [V_SEL, K_SEL*F]

    const int grid = (V_SEL + WAVES_PER_BLOCK - 1) / WAVES_PER_BLOCK;  // 6250
    knn_topk_gather_kernel<<<grid, BLOCK_THREADS, 0, stream>>>(
        feat, dist, nid, sel, out);
}